// ROIHeads_38903813767169
// MI455X (gfx1250) — compile-verified
//
#include <hip/hip_runtime.h>

#define BB 16
#define NN 4096
#define CC 91
#define NC1 90                 // C-1
#define CAND (NN * NC1)        // 368640 candidates per image
#define MSEL 4096              // top-M
#define KOUT 100               // NMS picks
#define W_CLIP 1333.0f
#define H_CLIP 800.0f
#define SCORE_TH 0.05f
#define NMS_TH 0.5f
#define LBL_OFF 10000.0f
#define DW_CLAMP 4.135166556742356f   // log(1000/16)

typedef __attribute__((ext_vector_type(2))) float v2f;
typedef __attribute__((ext_vector_type(8))) float v8f;
typedef __attribute__((ext_vector_type(4))) int  v4i;

#ifndef HAVE_ASYNC
#if defined(__has_builtin)
#if __has_builtin(__builtin_amdgcn_global_load_async_to_lds_b128)
#define HAVE_ASYNC 1
#endif
#endif
#endif
#ifndef HAVE_ASYNC
#define HAVE_ASYNC 0
#endif

#ifndef HAVE_WMMA4
#if defined(__has_builtin)
#if __has_builtin(__builtin_amdgcn_wmma_f32_16x16x4_f32)
#define HAVE_WMMA4 1
#endif
#endif
#endif
#ifndef HAVE_WMMA4
#define HAVE_WMMA4 0
#endif

__device__ __forceinline__ unsigned mono_key(float s) {
    unsigned u = __float_as_uint(s);
    return (u & 0x80000000u) ? ~u : (u | 0x80000000u);
}

// ---------------------------------------------------------------------------
// Kernel 1: softmax over C=91 per proposal row, threshold, write scores for
// classes 1..90. One wave processes 16 rows; block = 8 waves = 128 rows.
// Logits tile staged to LDS with CDNA5 async global->LDS loads; row-sum of
// exp computed with V_WMMA_F32_16X16X4_F32 (f32, exact) against a ones B.
// ---------------------------------------------------------------------------
__global__ __launch_bounds__(256) void softmax_scores_kernel(
    const float* __restrict__ logits, float* __restrict__ scores)
{
    __shared__ float lds[128 * 91];    // 46592 B
    __shared__ float sums[8 * 16];

    const int tid  = threadIdx.x;
    const int wv   = tid >> 5;
    const int lane = tid & 31;
    const int hi   = lane >> 4;        // half-wave: 0 or 1
    const int r    = lane & 15;        // row within 16-row tile
    const long long rowBase = (long long)blockIdx.x * 128;

    // ---- stage 128 rows = 11648 floats = 2912 float4 into LDS ----
    const float* src = logits + rowBase * 91;
#if HAVE_ASYNC
    #pragma unroll
    for (int i = 0; i < 12; ++i) {
        int idx = tid + i * 256;
        if (idx < 2912) {
            __builtin_amdgcn_global_load_async_to_lds_b128(
                (v4i*)(src + (size_t)idx * 4),
                (v4i*)(&lds[idx * 4]), 0, 0);
        }
    }
#if defined(__has_builtin) && __has_builtin(__builtin_amdgcn_s_wait_asynccnt)
    __builtin_amdgcn_s_wait_asynccnt(0);
#else
    asm volatile("s_wait_asynccnt 0" ::: "memory");
#endif
#else
    for (int i = tid; i < 2912; i += 256) {
        float4 v = ((const float4*)src)[i];
        ((float4*)lds)[i] = v;
    }
#endif
    __syncthreads();

    const float* myrow = &lds[(wv * 16 + r) * 91];

    // ---- row max (2 lanes per row, combine with xor-16 shuffle) ----
    float mx = -3.4e38f;
    {
        int k0 = hi ? 46 : 0, k1 = hi ? 91 : 46;
        for (int k = k0; k < k1; ++k) mx = fmaxf(mx, myrow[k]);
        mx = fmaxf(mx, __shfl_xor(mx, 16, 32));
    }

    // ---- sum of exp over 91 cols: 23 chunks of K=4 ----
    float rowSum;
#if HAVE_WMMA4
    {
        v8f acc = {0.f, 0.f, 0.f, 0.f, 0.f, 0.f, 0.f, 0.f};
        v2f bones; bones.x = 1.0f; bones.y = 1.0f;   // ones matrix: layout-invariant
        #pragma unroll
        for (int ch = 0; ch < 23; ++ch) {
            int k0 = ch * 4 + 2 * hi;                // A layout: hi=0 -> K0,K1 ; hi=1 -> K2,K3
            v2f a;
            a.x = (k0     < 91) ? __expf(myrow[k0]     - mx) : 0.0f;
            a.y = (k0 + 1 < 91) ? __expf(myrow[k0 + 1] - mx) : 0.0f;
            acc = __builtin_amdgcn_wmma_f32_16x16x4_f32(
                false, a, false, bones, (short)0, acc, false, false);
        }
        // D[m][n] = rowsum(m) for every n. Lane 0 holds rows 0..7 in acc[0..7],
        // lane 16 holds rows 8..15. N=0 lanes publish to LDS.
        if (r == 0) {
            #pragma unroll
            for (int v = 0; v < 8; ++v) sums[wv * 16 + v + 8 * hi] = acc[v];
        }
        __syncthreads();
        rowSum = sums[wv * 16 + r];
    }
#else
    {
        float part = 0.f;
        #pragma unroll
        for (int ch = 0; ch < 23; ++ch) {
            int k0 = ch * 4 + 2 * hi;
            if (k0     < 91) part += __expf(myrow[k0]     - mx);
            if (k0 + 1 < 91) part += __expf(myrow[k0 + 1] - mx);
        }
        rowSum = part + __shfl_xor(part, 16, 32);
        __syncthreads();
    }
#endif

    // ---- write thresholded scores for classes 1..90 ----
    float inv = 1.0f / rowSum;
    long long grow = rowBase + wv * 16 + r;
    float* dst = scores + grow * NC1;
    int c0 = hi ? 46 : 1, c1 = hi ? 91 : 46;
    for (int c = c0; c < c1; ++c) {
        float p = __expf(myrow[c] - mx) * inv;
        dst[c - 1] = (p > SCORE_TH) ? p : -1.0f;
    }
}

// ---------------------------------------------------------------------------
// Kernel 2: per-image top-M selection (12-bit radix histogram + deterministic
// prefix-scan compaction), then decode+clip boxes only for survivors.
// One 1024-thread workgroup per image.
// ---------------------------------------------------------------------------
__global__ __launch_bounds__(1024) void select_kernel(
    const float* __restrict__ scores,
    const float* __restrict__ boxreg,
    const float* __restrict__ props,
    float* __restrict__ topS, float* __restrict__ topB, float* __restrict__ topL)
{
    __shared__ unsigned hist[4096];          // 16 KB
    __shared__ unsigned cA[1024], cT[1024];  // per-thread counts -> exclusive offsets
    __shared__ int s_t, s_above, s_take;

    const int b   = blockIdx.x;
    const int tid = threadIdx.x;
    const float* sc = scores + (size_t)b * CAND;
    const int CHUNK = CAND / 1024;           // 360, exact

    for (int i = tid; i < 4096; i += 1024) hist[i] = 0;
    __syncthreads();

    // pass 1: histogram of top 12 bits of order-preserving key
    const int base = tid * CHUNK;
    for (int e = base; e < base + CHUNK; ++e) {
        __builtin_prefetch(sc + e + 64, 0, 0);
        atomicAdd(&hist[mono_key(sc[e]) >> 20], 1u);
    }
    __syncthreads();

    // find threshold bucket t (total CAND >= MSEL, so break always occurs)
    if (tid == 0) {
        unsigned cum = 0; int t = 0; int take = 0;
        for (int bin = 4095; bin >= 0; --bin) {
            unsigned c = hist[bin];
            if (cum + c >= MSEL) { t = bin; take = MSEL - (int)cum; break; }
            cum += c;
        }
        s_t = t; s_above = (int)cum; s_take = take;
    }
    __syncthreads();
    const int t = s_t, nAbove = s_above, take = s_take;

    // pass 2a: per-thread counts (deterministic, index-ordered)
    unsigned myA = 0, myT = 0;
    for (int e = base; e < base + CHUNK; ++e) {
        int bin = (int)(mono_key(sc[e]) >> 20);
        if (bin > t) ++myA; else if (bin == t) ++myT;
    }
    cA[tid] = myA; cT[tid] = myT;
    __syncthreads();
    if (tid == 0) {      // serial exclusive scan over 1024 (deterministic)
        unsigned a = 0, tt = 0;
        for (int i = 0; i < 1024; ++i) {
            unsigned ca = cA[i], ct = cT[i];
            cA[i] = a; cT[i] = tt;
            a += ca; tt += ct;
        }
    }
    __syncthreads();

    // pass 2b: assign slots in index order, decode + clip selected boxes
    unsigned posA = cA[tid], posT = cT[tid];
    for (int e = base; e < base + CHUNK; ++e) {
        float s = sc[e];
        int bin = (int)(mono_key(s) >> 20);
        int pos = -1;
        if (bin > t)       { pos = (int)posA++; }
        else if (bin == t) { unsigned j = posT++; if ((int)j < take) pos = nAbove + (int)j; }
        if (pos >= 0) {
            int n = e / NC1;
            int c = e - n * NC1 + 1;                      // class label 1..90
            size_t rr = ((size_t)(b * NN + n)) * (CC * 4) + (size_t)c * 4;
            float r0 = boxreg[rr + 0], r1 = boxreg[rr + 1];
            float r2 = boxreg[rr + 2], r3 = boxreg[rr + 3];
            const float* p = props + ((size_t)(b * NN + n)) * 4;
            float w  = p[2] - p[0], h = p[3] - p[1];
            float cx = p[0] + 0.5f * w, cy = p[1] + 0.5f * h;
            float dx = r0 * 0.1f, dy = r1 * 0.1f;
            float dw = fminf(r2 * 0.2f, DW_CLAMP);
            float dh = fminf(r3 * 0.2f, DW_CLAMP);
            float pcx = dx * w + cx, pcy = dy * h + cy;
            float pw  = __expf(dw) * w, ph = __expf(dh) * h;
            float x1 = fminf(fmaxf(pcx - 0.5f * pw, 0.f), W_CLIP);
            float y1 = fminf(fmaxf(pcy - 0.5f * ph, 0.f), H_CLIP);
            float x2 = fminf(fmaxf(pcx + 0.5f * pw, 0.f), W_CLIP);
            float y2 = fminf(fmaxf(pcy + 0.5f * ph, 0.f), H_CLIP);
            size_t o = (size_t)b * MSEL + pos;
            topS[o] = s;
            topB[o * 4 + 0] = x1; topB[o * 4 + 1] = y1;
            topB[o * 4 + 2] = x2; topB[o * 4 + 3] = y2;
            topL[o] = (float)c;
        }
    }
}

// ---------------------------------------------------------------------------
// Kernel 3: greedy NMS, one workgroup per image. Scores live in LDS; each
// thread caches 4 label-offset boxes + areas in registers. K=100 iterations
// of (block argmax with low-index tie-break) -> suppress.
// ---------------------------------------------------------------------------
__global__ __launch_bounds__(1024) void nms_kernel(
    const float* __restrict__ topS, const float* __restrict__ topB,
    const float* __restrict__ topL, float* __restrict__ out)
{
    __shared__ float sSc[MSEL];                  // 16 KB
    __shared__ unsigned long long red[32];
    __shared__ unsigned long long selKey;
    __shared__ float selBox[5];

    const int b    = blockIdx.x;
    const int tid  = threadIdx.x;
    const int wv   = tid >> 5;
    const int lane = tid & 31;

    float bx1[4], by1[4], bx2[4], by2[4], ar[4];
    #pragma unroll
    for (int j = 0; j < 4; ++j) {
        int i = tid + j * 1024;
        size_t o = (size_t)b * MSEL + i;
        float off = topL[o] * LBL_OFF;
        float x1 = topB[o * 4 + 0] + off, y1 = topB[o * 4 + 1] + off;
        float x2 = topB[o * 4 + 2] + off, y2 = topB[o * 4 + 3] + off;
        bx1[j] = x1; by1[j] = y1; bx2[j] = x2; by2[j] = y2;
        ar[j] = (x2 - x1) * (y2 - y1);
        sSc[i] = topS[o];
    }
    __syncthreads();

    for (int k = 0; k < KOUT; ++k) {
        // ---- block argmax, ties -> lowest index (matches jnp.argmax) ----
        unsigned long long best = 0;
        #pragma unroll
        for (int j = 0; j < 4; ++j) {
            int i = tid + j * 1024;
            unsigned long long kk =
                ((unsigned long long)mono_key(sSc[i]) << 32) |
                (unsigned)(0xFFFFFFFFu - (unsigned)i);
            best = best > kk ? best : kk;
        }
        #pragma unroll
        for (int off = 16; off; off >>= 1) {
            unsigned long long o2 = __shfl_xor(best, off, 32);
            best = best > o2 ? best : o2;
        }
        if (lane == 0) red[wv] = best;
        __syncthreads();
        if (wv == 0) {
            unsigned long long v = red[lane];
            #pragma unroll
            for (int off = 16; off; off >>= 1) {
                unsigned long long o2 = __shfl_xor(v, off, 32);
                v = v > o2 ? v : o2;
            }
            if (lane == 0) selKey = v;
        }
        __syncthreads();

        unsigned long long key = selKey;
        int i = (int)(0xFFFFFFFFu - (unsigned)key);
        unsigned mk = (unsigned)(key >> 32);                   // decode score from key
        unsigned su = (mk & 0x80000000u) ? (mk ^ 0x80000000u) : ~mk;
        float s = __uint_as_float(su);
        bool valid = s > 0.0f;

        if (tid == 0) {
            size_t o = (size_t)b * MSEL + i;
            float lbl = topL[o];
            float off = lbl * LBL_OFF;
            float x1 = topB[o * 4 + 0], y1 = topB[o * 4 + 1];
            float x2 = topB[o * 4 + 2], y2 = topB[o * 4 + 3];
            int oi = b * KOUT + k;
            out[oi * 4 + 0] = valid ? x1 : 0.f;                // boxes (B,K,4)
            out[oi * 4 + 1] = valid ? y1 : 0.f;
            out[oi * 4 + 2] = valid ? x2 : 0.f;
            out[oi * 4 + 3] = valid ? y2 : 0.f;
            out[BB * KOUT * 4 + oi] = valid ? s : 0.f;         // scores (B,K)
            out[BB * KOUT * 5 + oi] = valid ? lbl : -1.f;      // labels (B,K)
            out[BB * KOUT * 6 + oi] = valid ? 1.f : 0.f;       // valid  (B,K)
            selBox[0] = x1 + off; selBox[1] = y1 + off;
            selBox[2] = x2 + off; selBox[3] = y2 + off;
            selBox[4] = (x2 - x1) * (y2 - y1);
            sSc[i] = -1.0f;                                    // sc.at[i].set(-1)
        }
        __syncthreads();

        if (valid) {
            float sx1 = selBox[0], sy1 = selBox[1];
            float sx2 = selBox[2], sy2 = selBox[3], sa = selBox[4];
            #pragma unroll
            for (int j = 0; j < 4; ++j) {
                int i2 = tid + j * 1024;
                float iw = fmaxf(fminf(bx2[j], sx2) - fmaxf(bx1[j], sx1), 0.f);
                float ih = fmaxf(fminf(by2[j], sy2) - fmaxf(by1[j], sy1), 0.f);
                float inter = iw * ih;
                float iou = inter / (ar[j] + sa - inter + 1e-9f);
                if (iou > NMS_TH) sSc[i2] = -1.0f;
            }
        }
        __syncthreads();
    }
}

// ---------------------------------------------------------------------------
extern "C" void kernel_launch(void* const* d_in, const int* in_sizes, int n_in,
                              void* d_out, int out_size, void* d_ws, size_t ws_size,
                              hipStream_t stream) {
    const float* logits = (const float*)d_in[0];   // (B*N, 91)
    const float* boxreg = (const float*)d_in[1];   // (B*N, 364)
    const float* props  = (const float*)d_in[2];   // (B, N, 4)
    float* out = (float*)d_out;

    float* ws     = (float*)d_ws;
    float* scores = ws;                                     // B*CAND      (23.6 MB)
    float* topS   = scores + (size_t)BB * CAND;             // B*MSEL
    float* topB   = topS   + (size_t)BB * MSEL;             // B*MSEL*4
    float* topL   = topB   + (size_t)BB * MSEL * 4;         // B*MSEL

    softmax_scores_kernel<<<(BB * NN) / 128, 256, 0, stream>>>(logits, scores);
    select_kernel<<<BB, 1024, 0, stream>>>(scores, boxreg, props, topS, topB, topL);
    nms_kernel<<<BB, 1024, 0, stream>>>(topS, topB, topL, out);
}